// LightGCN_63591285785042
// MI455X (gfx1250) — compile-verified
//
#include <hip/hip_runtime.h>

#define N_AUTHORS 50000
#define N_PAPERS  100000
#define N_NODES   150000
#define N_EDGES   1200000
#define EMB       64

static constexpr unsigned NODE_F  = (unsigned)N_NODES * EMB;       // 9,600,000 floats
static constexpr unsigned NODE_F4 = NODE_F / 4;                    // 2,400,000 vec4
static constexpr unsigned AUTH_F4 = ((unsigned)N_AUTHORS * EMB)/4; // 800,000 vec4

#if defined(__has_builtin)
#if __has_builtin(__builtin_amdgcn_global_load_async_to_lds_b128)
#define HAVE_ASYNC_LDS 1
#endif
#endif

typedef int   v4i __attribute__((ext_vector_type(4)));
typedef float v4f __attribute__((ext_vector_type(4)));

// ---- concat(author, paper) -> ego region of d_out, and seed acc region ----
__global__ void k_init(const v4f* __restrict__ a,
                       const v4f* __restrict__ p,
                       v4f* __restrict__ ego,
                       v4f* __restrict__ acc) {
  unsigned i = blockIdx.x * blockDim.x + threadIdx.x;
  if (i >= NODE_F4) return;
  const v4f v = (i < AUTH_F4) ? a[i] : p[i - AUTH_F4];
  ego[i] = v;
  acc[i] = v;
}

__global__ void k_zero(v4f* __restrict__ y) {
  unsigned i = blockIdx.x * blockDim.x + threadIdx.x;
  if (i < NODE_F4) y[i] = (v4f)(0.f);
}

// acc = (acc + y) * s, and (optionally) zero the next layer's destination
// buffer in the same pass (zbuf may be nullptr).
__global__ void k_accum(const v4f* __restrict__ y,
                        v4f* __restrict__ acc,
                        v4f* __restrict__ zbuf,
                        float s) {
  unsigned i = blockIdx.x * blockDim.x + threadIdx.x;
  if (i >= NODE_F4) return;
  acc[i] = (acc[i] + y[i]) * s;
  if (zbuf) zbuf[i] = (v4f)(0.f);
}

// Edge-parallel SpMM scatter: 16 lanes per edge, one 128-bit gather per lane.
// Grid is exact: N_EDGES*16 threads / 256 = 75000 blocks (no tail, EXEC full).
__global__ void k_spmm(const int* __restrict__ rows,
                       const int* __restrict__ cols,
                       const float* __restrict__ vals,
                       const float* __restrict__ x,
                       float* __restrict__ y) {
  const unsigned t = blockIdx.x * blockDim.x + threadIdx.x;
  const unsigned e = t >> 4;        // edge id
  const unsigned q = t & 15u;       // vec4 slot within the 64-float row
  if (e >= (unsigned)N_EDGES) return;

  const int   r = rows[e];
  const int   c = cols[e];
  const float v = vals[e];

  const float* src = x + (size_t)c * EMB + q * 4u;

#ifdef HAVE_ASYNC_LDS
  // CDNA5 async-to-LDS gather (global_load_async_to_lds_b128, ASYNCcnt).
  __shared__ float smem[256 * 4];
  float* slot = &smem[(unsigned)threadIdx.x * 4u];
  __builtin_amdgcn_global_load_async_to_lds_b128((v4i*)src, (v4i*)slot, 0, 0);
#if __has_builtin(__builtin_amdgcn_s_wait_asynccnt)
  __builtin_amdgcn_s_wait_asynccnt(0);
#else
  asm volatile("s_wait_asynccnt 0" ::: "memory");
#endif
  const v4f m = *(const v4f*)slot;
#else
  const v4f m = *(const v4f*)src;          // global_load_b128
#endif

  float* dst = y + (size_t)r * EMB + q * 4u;
  // Native f32 atomics -> global_atomic_add_f32 (L2 RMW units)
  unsafeAtomicAdd(dst + 0, v * m.x);
  unsafeAtomicAdd(dst + 1, v * m.y);
  unsafeAtomicAdd(dst + 2, v * m.z);
  unsafeAtomicAdd(dst + 3, v * m.w);
}

extern "C" void kernel_launch(void* const* d_in, const int* in_sizes, int n_in,
                              void* d_out, int out_size, void* d_ws, size_t ws_size,
                              hipStream_t stream) {
  const float* auth = (const float*)d_in[0];
  const float* pap  = (const float*)d_in[1];
  const int*   rows = (const int*)d_in[2];
  const int*   cols = (const int*)d_in[3];
  const float* vals = (const float*)d_in[4];

  float* ego  = (float*)d_out;            // [0, 9.6M)   : ego
  float* acc  = ego + NODE_F;             // [9.6M,19.2M): acc -> (out_author|out_paper)
  float* bufA = (float*)d_ws;             // 38.4 MB
  float* bufB = bufA + NODE_F;            // 38.4 MB

  const dim3 blk(256);
  const dim3 gridN((NODE_F4 + 255u) / 256u);        // 9375 blocks
  const dim3 gridE((unsigned)N_EDGES * 16u / 256u); // 75000 blocks, exact

  k_init<<<gridN, blk, 0, stream>>>((const v4f*)auth, (const v4f*)pap,
                                    (v4f*)ego, (v4f*)acc);
  k_zero<<<gridN, blk, 0, stream>>>((v4f*)bufA);

  // layer 0: x=ego  -> y=bufA   (accum also zeroes bufB for layer 1)
  k_spmm<<<gridE, blk, 0, stream>>>(rows, cols, vals, ego, bufA);
  k_accum<<<gridN, blk, 0, stream>>>((const v4f*)bufA, (v4f*)acc, (v4f*)bufB, 1.0f);

  // layer 1: x=bufA -> y=bufB   (accum also zeroes bufA for layer 2)
  k_spmm<<<gridE, blk, 0, stream>>>(rows, cols, vals, bufA, bufB);
  k_accum<<<gridN, blk, 0, stream>>>((const v4f*)bufB, (v4f*)acc, (v4f*)bufA, 1.0f);

  // layer 2: x=bufB -> y=bufA   (final accum fuses the /4 scale)
  k_spmm<<<gridE, blk, 0, stream>>>(rows, cols, vals, bufB, bufA);
  k_accum<<<gridN, blk, 0, stream>>>((const v4f*)bufA, (v4f*)acc, (v4f*)nullptr, 0.25f);
}